// PWLUBase_4973572129165
// MI455X (gfx1250) — compile-verified
//
#include <hip/hip_runtime.h>
#include <stdint.h>

// PWLU (piecewise linear unit), channelwise, fixed bound ±2.3, 7 points / 6 regions.
// x: [B=64, C=256, H=56, W=56] f32; points: [C,7]; left_diffs/right_diffs: [C].
// Memory-bound streaming op: 411 MB traffic -> ~17.6 us floor at 23.3 TB/s.
//
// CDNA5 specifics used:
//  - global_load_async_to_lds_b32 (ASYNCcnt) to stage the per-channel table,
//  - s_wait_asynccnt 0,
//  - b128 non-temporal global loads/stores (write-once stream, 411MB > 192MB L2),
//  - ds_load_b64 gather for the 8-way per-lane (left_point, slope) select.

typedef float v4f __attribute__((ext_vector_type(4)));
typedef float v2f __attribute__((ext_vector_type(2)));

namespace {

constexpr int    kNP      = 7;          // breakpoints per channel
constexpr int    kHW      = 56 * 56;    // 3136 elements per (b,c) plane
constexpr int    kNVEC    = kHW / 4;    // 784 float4 per plane
constexpr int    kThreads = 256;        // 8 wave32 per block

__device__ __forceinline__ v2f mk2(float a, float b) {
  v2f t; t.x = a; t.y = b; return t;
}

// tab[r] = { left_point (false_points[r]), slope (diffs[r]) }, r in [0,7], in LDS.
__device__ __forceinline__ float pwlu_eval(float xv, const v2f* __restrict__ tab) {
  constexpr double kB         = 2.3;
  constexpr double kRL        = (2.0 * kB) / 6.0;
  constexpr float  kSimLeft   = (float)(-kB - kRL);      // -3.0666666f
  constexpr float  kInvRL     = (float)(1.0 / kRL);      // 1.3043478f
  constexpr float  kClipHi    = (float)(7 * 1.001);      // 7.0069999f
  // (x - sim_left) * (1/region_len): within 1 ulp of the reference's division;
  // the PWLU is continuous at breakpoints, so boundary rounding is benign.
  const float xn = (xv - kSimLeft) * kInvRL;
  const float xc = __builtin_fminf(__builtin_fmaxf(xn, 0.0f), kClipHi);
  const int   r  = (int)xc;                              // 0..7
  const v2f   t  = tab[r];                               // ds_load_b64 gather
  const float d  = xn - (float)r;
  return t.x + d * t.y;                                  // left_point + dist*slope
}

__global__ __launch_bounds__(kThreads) void pwlu_kernel(
    const float* __restrict__ x,
    const float* __restrict__ points,
    const float* __restrict__ ldiff,
    const float* __restrict__ rdiff,
    float* __restrict__ out) {
  __shared__ float s_raw[kNP + 2];   // p0..p6, left_diff, right_diff
  __shared__ v2f   s_tab[kNP + 1];   // 8 x {left_point, slope}

  const int c   = blockIdx.x;        // channel (block-uniform)
  const int tid = threadIdx.x;

  // --- Stage per-channel table inputs into LDS via gfx1250 async-to-LDS DMA.
  //     Branchless per-lane source select (v_cndmask on the 64-bit address).
  if (tid < kNP + 2) {
    const uint64_t ga_pts = (uint64_t)(uintptr_t)(points + c * kNP + tid);
    const uint64_t ga_ld  = (uint64_t)(uintptr_t)(ldiff + c);
    const uint64_t ga_rd  = (uint64_t)(uintptr_t)(rdiff + c);
    const uint64_t gaddr  = (tid < kNP) ? ga_pts : ((tid == kNP) ? ga_ld : ga_rd);
    const unsigned lds_off = (unsigned)(uintptr_t)&s_raw[tid];
    asm volatile("global_load_async_to_lds_b32 %0, %1, off"
                 :: "v"(lds_off), "v"(gaddr)
                 : "memory");
  }
  asm volatile("s_wait_asynccnt 0" ::: "memory");  // ASYNCcnt==0: LDS written
  __syncthreads();

  // --- Build the (left_point, slope) table: one lane per region, no serial chain.
  if (tid < kNP + 1) {                 // tid = r in [0,7]
    float lp, sl;
    if (tid == 0)            { lp = s_raw[0] - s_raw[kNP]; sl = s_raw[kNP];     }
    else if (tid == kNP)     { lp = s_raw[kNP - 1];        sl = s_raw[kNP + 1]; }
    else                     { lp = s_raw[tid - 1];        sl = s_raw[tid] - s_raw[tid - 1]; }
    s_tab[tid] = mk2(lp, sl);
  }
  __syncthreads();

  // --- Stream one (b,c) plane: 784 float4 per block, NT b128 loads/stores.
  const size_t base =
      ((size_t)blockIdx.y * (size_t)gridDim.x + (size_t)c) * (size_t)kHW;
  const v4f* __restrict__ xv = reinterpret_cast<const v4f*>(x + base);
  v4f* __restrict__ ov       = reinterpret_cast<v4f*>(out + base);

#pragma unroll
  for (int k = 0; k < kNVEC / kThreads; ++k) {   // 3 full sweeps
    const int i = tid + k * kThreads;
    v4f v = __builtin_nontemporal_load(xv + i);
    v4f o;
    o.x = pwlu_eval(v.x, s_tab);
    o.y = pwlu_eval(v.y, s_tab);
    o.z = pwlu_eval(v.z, s_tab);
    o.w = pwlu_eval(v.w, s_tab);
    __builtin_nontemporal_store(o, ov + i);
  }
  {                                              // tail: 784 % 256 = 16 vectors
    const int i = tid + (kNVEC / kThreads) * kThreads;
    if (i < kNVEC) {
      v4f v = __builtin_nontemporal_load(xv + i);
      v4f o;
      o.x = pwlu_eval(v.x, s_tab);
      o.y = pwlu_eval(v.y, s_tab);
      o.z = pwlu_eval(v.z, s_tab);
      o.w = pwlu_eval(v.w, s_tab);
      __builtin_nontemporal_store(o, ov + i);
    }
  }
}

} // namespace

extern "C" void kernel_launch(void* const* d_in, const int* in_sizes, int n_in,
                              void* d_out, int out_size, void* d_ws, size_t ws_size,
                              hipStream_t stream) {
  const float* x      = (const float*)d_in[0];
  const float* points = (const float*)d_in[1];
  const float* ld     = (const float*)d_in[2];
  const float* rd     = (const float*)d_in[3];
  float*       out    = (float*)d_out;

  const int C = in_sizes[2];                                    // 256 (left_diffs)
  const long long total = (long long)out_size;                  // B*C*H*W
  const int B = (int)(total / ((long long)C * (long long)kHW)); // 64

  dim3 grid((unsigned)C, (unsigned)B, 1);
  pwlu_kernel<<<grid, dim3(kThreads), 0, stream>>>(x, points, ld, rd, out);
}